// Sparsemax_19198503813450
// MI455X (gfx1250) — compile-verified
//
#include <hip/hip_runtime.h>
#include <cstdint>

#define D      4096
#define TPB    256
#define NWAVE  (TPB / 32)
#define CHUNKS 4          // 4 x float4 per thread = 16 values
#define VPT    16
#define NITER  16

typedef float v4f __attribute__((ext_vector_type(4)));

__device__ __forceinline__ float wave_max(float v) {
#pragma unroll
    for (int off = 16; off > 0; off >>= 1)
        v = fmaxf(v, __shfl_xor(v, off, 32));
    return v;
}

__device__ __forceinline__ float wave_sum(float v) {
#pragma unroll
    for (int off = 16; off > 0; off >>= 1)
        v += __shfl_xor(v, off, 32);
    return v;
}

__global__ __launch_bounds__(TPB) void sparsemax_rows_kernel(
        const float* __restrict__ x, float* __restrict__ y) {
    __shared__ __align__(16) float smem[D];
    __shared__ float rbuf[NWAVE];
    __shared__ float rbuf2[NWAVE];
    __shared__ float bval;
    __shared__ float bval2;

    const int tid  = threadIdx.x;
    const int wave = tid >> 5;
    const int lane = tid & 31;

    const long long row = blockIdx.x;
    const float* __restrict__ xrow = x + row * (long long)D;
    float* __restrict__ yrow       = y + row * (long long)D;

    // ---- CDNA5 async copy: HBM -> LDS, bypassing VGPRs (ASYNCcnt path). ----
    // Streaming input: non-temporal hint so the one-shot row doesn't churn L2.
#pragma unroll
    for (int c = 0; c < CHUNKS; ++c) {
        const int e = (c * TPB + tid) * 4;                    // float4 element index
        const unsigned long long gaddr =
            (unsigned long long)(uintptr_t)(xrow + e);
        const unsigned lds_off = (unsigned)(uintptr_t)(&smem[e]); // low 32b = LDS byte addr
        asm volatile("global_load_async_to_lds_b128 %0, %1, off th:TH_LOAD_NT"
                     :: "v"(lds_off), "v"(gaddr)
                     : "memory");
    }
    asm volatile("s_wait_asynccnt 0" ::: "memory");
    __syncthreads();

    // ---- pull my 16 values into registers ----
    float v[VPT];
#pragma unroll
    for (int c = 0; c < CHUNKS; ++c) {
        const int e = (c * TPB + tid) * 4;
        const v4f q = *(const v4f*)&smem[e];
        v[c * 4 + 0] = q.x;
        v[c * 4 + 1] = q.y;
        v[c * 4 + 2] = q.z;
        v[c * 4 + 3] = q.w;
    }

    // ---- fused block-wide (max, sum) reduction in one barrier round ----
    float m = v[0];
    float s = v[0];
#pragma unroll
    for (int i = 1; i < VPT; ++i) { m = fmaxf(m, v[i]); s += v[i]; }
    m = wave_max(m);
    s = wave_sum(s);
    if (lane == 0) { rbuf[wave] = m; rbuf2[wave] = s; }
    __syncthreads();
    if (tid < 32) {
        float tm = (lane < NWAVE) ? rbuf[lane]  : -__builtin_inff();
        float ts = (lane < NWAVE) ? rbuf2[lane] : 0.f;
        tm = wave_max(tm);
        ts = wave_sum(ts);
        if (lane == 0) { bval = tm; bval2 = ts; }
    }
    __syncthreads();
    const float zmax = bval;
    // shifted-row sum: sum(x - zmax) = sum(x) - D*zmax
    float tau = (bval2 - (float)D * zmax - 1.0f) * (1.0f / (float)D);

    // ---- shift values once in registers ----
#pragma unroll
    for (int i = 0; i < VPT; ++i) v[i] -= zmax;

    // ---- Michelot fixed-point: tau <- (sum_{z>tau} z - 1) / |{z>tau}| ----
    // tau increases monotonically to the exact sparsemax threshold.
    // k >= 1 always (shifted max z=0 and tau < 0 throughout).
    // Convergence test is block-uniform (all threads read the same bval).
    for (int it = 0; it < NITER; ++it) {
        float ps = 0.f, pk = 0.f;
#pragma unroll
        for (int i = 0; i < VPT; ++i) {
            if (v[i] > tau) { ps += v[i]; pk += 1.0f; }
        }
        ps = wave_sum(ps);
        pk = wave_sum(pk);
        if (lane == 0) { rbuf[wave] = ps; rbuf2[wave] = pk; }
        __syncthreads();
        if (tid < 32) {
            float ts = (lane < NWAVE) ? rbuf[lane]  : 0.f;
            float tk = (lane < NWAVE) ? rbuf2[lane] : 0.f;
            ts = wave_sum(ts);
            tk = wave_sum(tk);
            if (lane == 0) bval = (ts - 1.0f) / tk;
        }
        __syncthreads();
        const float ntau = bval;
        if (ntau == tau) break;   // uniform: all threads see identical bval
        tau = ntau;
    }

    // ---- emit: max(z - tau, 0); write-once stream -> non-temporal stores ----
#pragma unroll
    for (int c = 0; c < CHUNKS; ++c) {
        const int e = (c * TPB + tid) * 4;
        v4f o;
        o.x = fmaxf(v[c * 4 + 0] - tau, 0.f);
        o.y = fmaxf(v[c * 4 + 1] - tau, 0.f);
        o.z = fmaxf(v[c * 4 + 2] - tau, 0.f);
        o.w = fmaxf(v[c * 4 + 3] - tau, 0.f);
        __builtin_nontemporal_store(o, (v4f*)&yrow[e]);
    }
}

extern "C" void kernel_launch(void* const* d_in, const int* in_sizes, int n_in,
                              void* d_out, int out_size, void* d_ws, size_t ws_size,
                              hipStream_t stream) {
    (void)n_in; (void)out_size; (void)d_ws; (void)ws_size;
    const float* x = (const float*)d_in[0];
    float* y = (float*)d_out;
    const long long n    = (long long)in_sizes[0];
    const int       rows = (int)(n / D);   // 16*2048 = 32768 rows of 4096
    sparsemax_rows_kernel<<<rows, TPB, 0, stream>>>(x, y);
}